// ResidualBlock_51616916963478
// MI455X (gfx1250) — compile-verified
//
#include <hip/hip_runtime.h>
#include <math.h>

// ---------------------------------------------------------------------------
// Problem constants (from the reference)
// ---------------------------------------------------------------------------
#define BSZ   512
#define CCH   256
#define LSQ   100
#define HCH   512     // 2*C
#define NST   64
#define SIDE_  72
#define DIFF_ 128
#define LT32    4     // ceil(L/32) 32-wide output column blocks

typedef __attribute__((ext_vector_type(2))) float v2f;
typedef __attribute__((ext_vector_type(8))) float v8f;

#define V8F_ZERO {0.f,0.f,0.f,0.f,0.f,0.f,0.f,0.f}

// D = A(16x4,f32) * B(4x16,f32) + C(16x16,f32), full fp32 precision.
__device__ __forceinline__ v8f wmma4(v2f a, v2f b, v8f c) {
  return __builtin_amdgcn_wmma_f32_16x16x4_f32(
      /*neg_a=*/false, a, /*neg_b=*/false, b,
      /*c_mod=*/(short)0, c, /*reuse_a=*/false, /*reuse_b=*/false);
}

// ---------------------------------------------------------------------------
// 1) d = diffusion_emb @ W_dp^T + b_dp            [B,C]
//    grid (B/16, C/16), 1 wave / tile, K = 128
// ---------------------------------------------------------------------------
__global__ void diff_proj_wmma(const float* __restrict__ emb,
                               const float* __restrict__ Wdp,
                               const float* __restrict__ bdp,
                               float* __restrict__ dvec) {
  const int m0 = blockIdx.x * 16, n0 = blockIdx.y * 16;
  const int lane = threadIdx.x & 31, half = lane >> 4, lid = lane & 15;
  v8f acc = V8F_ZERO;
  for (int k0 = 0; k0 < DIFF_; k0 += 4) {
    const int ka = k0 + 2 * half;
    v2f a, b;
    a.x = emb[(m0 + lid) * DIFF_ + ka];
    a.y = emb[(m0 + lid) * DIFF_ + ka + 1];
    b.x = Wdp[(n0 + lid) * DIFF_ + ka];       // B[k,n] = W_dp[n,k]
    b.y = Wdp[(n0 + lid) * DIFF_ + ka + 1];
    acc = wmma4(a, b, acc);
  }
  for (int r = 0; r < 8; ++r) {
    const int row = m0 + r + 8 * half, col = n0 + lid;
    dvec[row * CCH + col] = acc[r] + bdp[col];
  }
}

// ---------------------------------------------------------------------------
// 2) y0 = x + d[b,c]  (broadcast over L)
// ---------------------------------------------------------------------------
__global__ void add_diff_kernel(const float* __restrict__ x,
                                const float* __restrict__ dvec,
                                float* __restrict__ y0) {
  const size_t i = (size_t)blockIdx.x * blockDim.x + threadIdx.x;
  if (i >= (size_t)BSZ * CCH * LSQ) return;
  const size_t bc = i / LSQ;
  y0[i] = x[i] + dvec[bc];
}

// ---------------------------------------------------------------------------
// 3a) per-(h,n) discretization:  dtA, Cb = C*(exp(dtA)-1)/A
// ---------------------------------------------------------------------------
__global__ void s4_cb_kernel(const float* __restrict__ log_dt,
                             const float* __restrict__ logA,
                             const float* __restrict__ Aim,
                             const float* __restrict__ Cre,
                             const float* __restrict__ Cim,
                             float* __restrict__ dtar, float* __restrict__ dtai,
                             float* __restrict__ cbr,  float* __restrict__ cbi) {
  const int i = blockIdx.x * blockDim.x + threadIdx.x;
  if (i >= HCH * NST) return;
  const int h  = i / NST;
  const float dt = expf(log_dt[h]);
  const float ar = -expf(logA[i]);
  const float ai = Aim[i];
  const float dr = ar * dt, di = ai * dt;
  const float er = expf(dr);
  const float ec = er * cosf(di), es = er * sinf(di);
  const float mr = ec - 1.f, mi = es;                 // exp(dtA)-1
  const float cr = Cre[i],  ci = Cim[i];
  const float tr = cr * mr - ci * mi;                 // C*(exp(dtA)-1)
  const float ti = cr * mi + ci * mr;
  const float den = ar * ar + ai * ai;                // / A
  cbr[i] = (tr * ar + ti * ai) / den;
  cbi[i] = (ti * ar - tr * ai) / den;
  dtar[i] = dr; dtai[i] = di;
}

// 3b) K[h,l] = 2*Re( sum_n Cb * exp(dtA*l) )
__global__ void s4_gen_kernel(const float* __restrict__ dtar,
                              const float* __restrict__ dtai,
                              const float* __restrict__ cbr,
                              const float* __restrict__ cbi,
                              float* __restrict__ K) {
  const int i = blockIdx.x * blockDim.x + threadIdx.x;
  if (i >= HCH * LSQ) return;
  const int h = i / LSQ, l = i % LSQ;
  const float fl = (float)l;
  const float* pr = dtar + h * NST;
  const float* pi = dtai + h * NST;
  const float* br = cbr  + h * NST;
  const float* bi = cbi  + h * NST;
  float acc = 0.f;
  for (int n = 0; n < NST; ++n) {
    const float er = expf(pr[n] * fl);
    const float ph = pi[n] * fl;
    acc += er * (br[n] * cosf(ph) - bi[n] * sinf(ph));
  }
  K[i] = 2.f * acc;
}

// ---------------------------------------------------------------------------
// 4) Conv1d k=3 pad=1, C->H, WMMA GEMM over (c,t), 2x2 register tiling.
//    Wave computes 32h x 32l.  grid (LT32, H/32, B)
// ---------------------------------------------------------------------------
__global__ void conv3_wmma(const float* __restrict__ y0,
                           const float* __restrict__ Wc,
                           const float* __restrict__ bc,
                           float* __restrict__ y1) {
  const int l0 = blockIdx.x * 32, h0 = blockIdx.y * 32, bi = blockIdx.z;
  const int lane = threadIdx.x & 31, half = lane >> 4, lid = lane & 15;
  const float* ub = y0 + (size_t)bi * CCH * LSQ;
  v8f a00 = V8F_ZERO, a01 = V8F_ZERO, a10 = V8F_ZERO, a11 = V8F_ZERO;
  for (int t = 0; t < 3; ++t) {
    const int ls0 = l0 + lid + t - 1;                 // shifted source column
    const int ls1 = ls0 + 16;
    const bool ok0 = (ls0 >= 0) && (ls0 < LSQ);
    const bool ok1 = (ls1 >= 0) && (ls1 < LSQ);
    for (int c0 = 0; c0 < CCH; c0 += 4) {
      const int ka = c0 + 2 * half;
      v2f fa0, fa1, fb0, fb1;
      fa0.x = Wc[((h0      + lid) * CCH + ka    ) * 3 + t];
      fa0.y = Wc[((h0      + lid) * CCH + ka + 1) * 3 + t];
      fa1.x = Wc[((h0 + 16 + lid) * CCH + ka    ) * 3 + t];
      fa1.y = Wc[((h0 + 16 + lid) * CCH + ka + 1) * 3 + t];
      fb0.x = ok0 ? ub[(size_t)(ka    ) * LSQ + ls0] : 0.f;
      fb0.y = ok0 ? ub[(size_t)(ka + 1) * LSQ + ls0] : 0.f;
      fb1.x = ok1 ? ub[(size_t)(ka    ) * LSQ + ls1] : 0.f;
      fb1.y = ok1 ? ub[(size_t)(ka + 1) * LSQ + ls1] : 0.f;
      a00 = wmma4(fa0, fb0, a00);
      a01 = wmma4(fa0, fb1, a01);
      a10 = wmma4(fa1, fb0, a10);
      a11 = wmma4(fa1, fb1, a11);
    }
  }
  const int lA = l0 + lid, lB = lA + 16;
  for (int r = 0; r < 8; ++r) {
    const int hA = h0 + r + 8 * half, hB = hA + 16;
    if (lA < LSQ) {
      y1[((size_t)bi * HCH + hA) * LSQ + lA] = a00[r] + bc[hA];
      y1[((size_t)bi * HCH + hB) * LSQ + lA] = a10[r] + bc[hB];
    }
    if (lB < LSQ) {
      y1[((size_t)bi * HCH + hA) * LSQ + lB] = a01[r] + bc[hA];
      y1[((size_t)bi * HCH + hB) * LSQ + lB] = a11[r] + bc[hB];
    }
  }
}

// ---------------------------------------------------------------------------
// 5/8) S4 causal Toeplitz conv + D*u + residual:  z = u*(1+D) + K (*) u
//      Per-h GEMM [B x L]x[L x L] (lower triangular), 2x2 tiling,
//      kernel row staged in LDS.  grid (LT32, B/32, H)
// ---------------------------------------------------------------------------
__global__ void s4conv_wmma(const float* __restrict__ u,
                            const float* __restrict__ K,
                            const float* __restrict__ Dp,
                            float* __restrict__ z) {
  const int l0 = blockIdx.x * 32, b0 = blockIdx.y * 32, h = blockIdx.z;
  const int lane = threadIdx.x & 31, half = lane >> 4, lid = lane & 15;
  __shared__ float sK[LSQ];
  for (int i = lane; i < LSQ; i += 32) sK[i] = K[(size_t)h * LSQ + i];
  __syncthreads();
  v8f a00 = V8F_ZERO, a01 = V8F_ZERO, a10 = V8F_ZERO, a11 = V8F_ZERO;
  const int la = l0 + lid, lb = la + 16;
  const int mEnd = (l0 + 32 < LSQ) ? (l0 + 32) : LSQ; // skip zero upper tiles
  for (int m0 = 0; m0 < mEnd; m0 += 4) {
    const int ka = m0 + 2 * half;
    const float* up0 = u + ((size_t)(b0      + lid) * HCH + h) * LSQ;
    const float* up1 = u + ((size_t)(b0 + 16 + lid) * HCH + h) * LSQ;
    v2f fa0, fa1, fb0, fb1;
    fa0.x = (ka     < LSQ) ? up0[ka]     : 0.f;        // A[b, m]
    fa0.y = (ka + 1 < LSQ) ? up0[ka + 1] : 0.f;
    fa1.x = (ka     < LSQ) ? up1[ka]     : 0.f;
    fa1.y = (ka + 1 < LSQ) ? up1[ka + 1] : 0.f;
    fb0.x = (la < LSQ && ka     <= la) ? sK[la - ka]     : 0.f;  // B[m,l]=K[l-m]
    fb0.y = (la < LSQ && ka + 1 <= la) ? sK[la - ka - 1] : 0.f;
    fb1.x = (lb < LSQ && ka     <= lb) ? sK[lb - ka]     : 0.f;
    fb1.y = (lb < LSQ && ka + 1 <= lb) ? sK[lb - ka - 1] : 0.f;
    a00 = wmma4(fa0, fb0, a00);
    a01 = wmma4(fa0, fb1, a01);
    a10 = wmma4(fa1, fb0, a10);
    a11 = wmma4(fa1, fb1, a11);
  }
  const float dh = Dp[h];
  for (int r = 0; r < 8; ++r) {
    const int bA = b0 + r + 8 * half, bB = bA + 16;
    if (la < LSQ) {
      const size_t i0 = ((size_t)bA * HCH + h) * LSQ + la;
      const size_t i1 = ((size_t)bB * HCH + h) * LSQ + la;
      z[i0] = u[i0] * (1.f + dh) + a00[r];
      z[i1] = u[i1] * (1.f + dh) + a10[r];
    }
    if (lb < LSQ) {
      const size_t i0 = ((size_t)bA * HCH + h) * LSQ + lb;
      const size_t i1 = ((size_t)bB * HCH + h) * LSQ + lb;
      z[i0] = u[i0] * (1.f + dh) + a01[r];
      z[i1] = u[i1] * (1.f + dh) + a11[r];
    }
  }
}

// ---------------------------------------------------------------------------
// 6/9) LayerNorm over channel dim H=512, per (b,l).  In place.
// ---------------------------------------------------------------------------
__global__ void layernorm_kernel(float* __restrict__ z,
                                 const float* __restrict__ g,
                                 const float* __restrict__ beta) {
  const int bl = blockIdx.x, b = bl / LSQ, l = bl % LSQ;
  const int tid = threadIdx.x;                        // blockDim = 256
  __shared__ float ssum[256], ssq[256];
  float* zp = z + (size_t)b * HCH * LSQ + l;
  const float v0 = zp[(size_t)tid * LSQ];
  const float v1 = zp[(size_t)(tid + 256) * LSQ];
  ssum[tid] = v0 + v1;
  ssq[tid]  = v0 * v0 + v1 * v1;
  __syncthreads();
  for (int s = 128; s > 0; s >>= 1) {
    if (tid < s) { ssum[tid] += ssum[tid + s]; ssq[tid] += ssq[tid + s]; }
    __syncthreads();
  }
  const float mu  = ssum[0] * (1.f / HCH);
  const float var = ssq[0] * (1.f / HCH) - mu * mu;
  const float inv = rsqrtf(var + 1e-5f);
  zp[(size_t)tid * LSQ]         = g[tid]       * (v0 - mu) * inv + beta[tid];
  zp[(size_t)(tid + 256) * LSQ] = g[tid + 256] * (v1 - mu) * inv + beta[tid + 256];
}

// ---------------------------------------------------------------------------
// 7) y += W_cond@[cond_obs;cond_mask] + W_time@time_emb + biases (in place)
//    2x2 tiling.  grid (LT32, H/32, B).  K = 144 then 128.
// ---------------------------------------------------------------------------
__global__ void cond_add_wmma(float* __restrict__ y,
                              const float* __restrict__ cond_obs,
                              const float* __restrict__ cond_mask,
                              const float* __restrict__ time_emb,
                              const float* __restrict__ Wcond,
                              const float* __restrict__ bcond,
                              const float* __restrict__ Wtime,
                              const float* __restrict__ btime) {
  const int l0 = blockIdx.x * 32, h0 = blockIdx.y * 32, bi = blockIdx.z;
  const int lane = threadIdx.x & 31, half = lane >> 4, lid = lane & 15;
  const int la = l0 + lid, lb = la + 16;
  const bool ok0 = la < LSQ, ok1 = lb < LSQ;
  const int lc0 = ok0 ? la : 0, lc1 = ok1 ? lb : 0;
  v8f a00 = V8F_ZERO, a01 = V8F_ZERO, a10 = V8F_ZERO, a11 = V8F_ZERO;
  // side conditioning, K = 2*SIDE = 144
  for (int k0 = 0; k0 < 2 * SIDE_; k0 += 4) {
    const int ka = k0 + 2 * half;
    v2f fa0, fa1, fb0, fb1;
    fa0.x = Wcond[(h0      + lid) * (2 * SIDE_) + ka];
    fa0.y = Wcond[(h0      + lid) * (2 * SIDE_) + ka + 1];
    fa1.x = Wcond[(h0 + 16 + lid) * (2 * SIDE_) + ka];
    fa1.y = Wcond[(h0 + 16 + lid) * (2 * SIDE_) + ka + 1];
    const size_t r0 = (ka     < SIDE_) ? ((size_t)bi * SIDE_ + ka        ) * LSQ
                                       : (size_t)0;
    const float* s0p = (ka     < SIDE_) ? cond_obs  + ((size_t)bi * SIDE_ + ka            ) * LSQ
                                        : cond_mask + ((size_t)bi * SIDE_ + (ka - SIDE_)  ) * LSQ;
    const float* s1p = (ka + 1 < SIDE_) ? cond_obs  + ((size_t)bi * SIDE_ + ka + 1        ) * LSQ
                                        : cond_mask + ((size_t)bi * SIDE_ + (ka + 1 - SIDE_)) * LSQ;
    (void)r0;
    fb0.x = ok0 ? s0p[lc0] : 0.f;
    fb0.y = ok0 ? s1p[lc0] : 0.f;
    fb1.x = ok1 ? s0p[lc1] : 0.f;
    fb1.y = ok1 ? s1p[lc1] : 0.f;
    a00 = wmma4(fa0, fb0, a00);
    a01 = wmma4(fa0, fb1, a01);
    a10 = wmma4(fa1, fb0, a10);
    a11 = wmma4(fa1, fb1, a11);
  }
  // time embedding, K = 128
  for (int k0 = 0; k0 < 128; k0 += 4) {
    const int ka = k0 + 2 * half;
    v2f fa0, fa1, fb0, fb1;
    fa0.x = Wtime[(h0      + lid) * 128 + ka];
    fa0.y = Wtime[(h0      + lid) * 128 + ka + 1];
    fa1.x = Wtime[(h0 + 16 + lid) * 128 + ka];
    fa1.y = Wtime[(h0 + 16 + lid) * 128 + ka + 1];
    const float* t0p = time_emb + ((size_t)bi * 128 + ka    ) * LSQ;
    const float* t1p = time_emb + ((size_t)bi * 128 + ka + 1) * LSQ;
    fb0.x = ok0 ? t0p[lc0] : 0.f;
    fb0.y = ok0 ? t1p[lc0] : 0.f;
    fb1.x = ok1 ? t0p[lc1] : 0.f;
    fb1.y = ok1 ? t1p[lc1] : 0.f;
    a00 = wmma4(fa0, fb0, a00);
    a01 = wmma4(fa0, fb1, a01);
    a10 = wmma4(fa1, fb0, a10);
    a11 = wmma4(fa1, fb1, a11);
  }
  for (int r = 0; r < 8; ++r) {
    const int hA = h0 + r + 8 * half, hB = hA + 16;
    if (ok0) {
      const size_t i0 = ((size_t)bi * HCH + hA) * LSQ + la;
      const size_t i1 = ((size_t)bi * HCH + hB) * LSQ + la;
      y[i0] = y[i0] + a00[r] + bcond[hA] + btime[hA];
      y[i1] = y[i1] + a10[r] + bcond[hB] + btime[hB];
    }
    if (ok1) {
      const size_t i0 = ((size_t)bi * HCH + hA) * LSQ + lb;
      const size_t i1 = ((size_t)bi * HCH + hB) * LSQ + lb;
      y[i0] = y[i0] + a01[r] + bcond[hA] + btime[hA];
      y[i1] = y[i1] + a11[r] + bcond[hB] + btime[hB];
    }
  }
}

// ---------------------------------------------------------------------------
// 10) gated = sigmoid(y[:, :C]) * tanh(y[:, C:])    [B,C,L]
// ---------------------------------------------------------------------------
__global__ void gate_kernel(const float* __restrict__ y, float* __restrict__ g) {
  const size_t i = (size_t)blockIdx.x * blockDim.x + threadIdx.x;
  if (i >= (size_t)BSZ * CCH * LSQ) return;
  const size_t b  = i / ((size_t)CCH * LSQ);
  const size_t cl = i % ((size_t)CCH * LSQ);
  const float ga = y[(size_t)b * HCH * LSQ + cl];
  const float fi = y[(size_t)b * HCH * LSQ + (size_t)CCH * LSQ + cl];
  g[i] = (1.f / (1.f + expf(-ga))) * tanhf(fi);
}

// ---------------------------------------------------------------------------
// 11) out = W_out @ gated + b_out; split -> (x+res)/sqrt2 | skip
//     2x2 tiling.  grid (LT32, H/32, B).  K = 256.
// ---------------------------------------------------------------------------
__global__ void out_proj_wmma(const float* __restrict__ gbuf,
                              const float* __restrict__ Wout,
                              const float* __restrict__ bout,
                              const float* __restrict__ x,
                              float* __restrict__ out) {
  const int l0 = blockIdx.x * 32, o0 = blockIdx.y * 32, bi = blockIdx.z;
  const int lane = threadIdx.x & 31, half = lane >> 4, lid = lane & 15;
  const int la = l0 + lid, lb = la + 16;
  const bool ok0 = la < LSQ, ok1 = lb < LSQ;
  const int lc0 = ok0 ? la : 0, lc1 = ok1 ? lb : 0;
  v8f a00 = V8F_ZERO, a01 = V8F_ZERO, a10 = V8F_ZERO, a11 = V8F_ZERO;
  for (int k0 = 0; k0 < CCH; k0 += 4) {
    const int ka = k0 + 2 * half;
    v2f fa0, fa1, fb0, fb1;
    fa0.x = Wout[(o0      + lid) * CCH + ka];
    fa0.y = Wout[(o0      + lid) * CCH + ka + 1];
    fa1.x = Wout[(o0 + 16 + lid) * CCH + ka];
    fa1.y = Wout[(o0 + 16 + lid) * CCH + ka + 1];
    const float* g0 = gbuf + ((size_t)bi * CCH + ka    ) * LSQ;
    const float* g1 = gbuf + ((size_t)bi * CCH + ka + 1) * LSQ;
    fb0.x = ok0 ? g0[lc0] : 0.f;
    fb0.y = ok0 ? g1[lc0] : 0.f;
    fb1.x = ok1 ? g0[lc1] : 0.f;
    fb1.y = ok1 ? g1[lc1] : 0.f;
    a00 = wmma4(fa0, fb0, a00);
    a01 = wmma4(fa0, fb1, a01);
    a10 = wmma4(fa1, fb0, a10);
    a11 = wmma4(fa1, fb1, a11);
  }
  const float inv_s2 = 0.70710678118654752440f;
  for (int r = 0; r < 8; ++r) {
    const int oA = o0 + r + 8 * half, oB = oA + 16;
    for (int half_t = 0; half_t < 2; ++half_t) {
      const int  l  = half_t ? lb : la;
      const bool ok = half_t ? ok1 : ok0;
      if (!ok) continue;
      const float vA = (half_t ? a01[r] : a00[r]) + bout[oA];
      const float vB = (half_t ? a11[r] : a10[r]) + bout[oB];
      // tile A rows
      if (oA < CCH) {
        const size_t xi = ((size_t)bi * CCH + oA) * LSQ + l;
        out[xi] = (x[xi] + vA) * inv_s2;
      } else {
        out[(size_t)BSZ * CCH * LSQ + ((size_t)bi * CCH + (oA - CCH)) * LSQ + l] = vA;
      }
      // tile B rows
      if (oB < CCH) {
        const size_t xi = ((size_t)bi * CCH + oB) * LSQ + l;
        out[xi] = (x[xi] + vB) * inv_s2;
      } else {
        out[(size_t)BSZ * CCH * LSQ + ((size_t)bi * CCH + (oB - CCH)) * LSQ + l] = vB;
      }
    }
  }
}

// ---------------------------------------------------------------------------
// Launch pipeline
// ---------------------------------------------------------------------------
extern "C" void kernel_launch(void* const* d_in, const int* in_sizes, int n_in,
                              void* d_out, int out_size, void* d_ws, size_t ws_size,
                              hipStream_t stream) {
  const float* x         = (const float*)d_in[0];
  const float* cond_obs  = (const float*)d_in[1];
  const float* time_emb  = (const float*)d_in[2];
  /* d_in[3] feature_emb is unused by the reference */
  const float* cond_mask = (const float*)d_in[4];
  const float* diff_emb  = (const float*)d_in[5];
  const float* W_dp      = (const float*)d_in[6];
  const float* b_dp      = (const float*)d_in[7];
  const float* W_conv    = (const float*)d_in[8];
  const float* b_conv    = (const float*)d_in[9];
  const float* W_cond    = (const float*)d_in[10];
  const float* b_cond    = (const float*)d_in[11];
  const float* W_time    = (const float*)d_in[12];
  const float* b_time    = (const float*)d_in[13];
  const float* W_out     = (const float*)d_in[14];
  const float* b_out     = (const float*)d_in[15];
  const float* s4a[8];  for (int i = 0; i < 8; ++i) s4a[i] = (const float*)d_in[16 + i];
  const float* s4b[8];  for (int i = 0; i < 8; ++i) s4b[i] = (const float*)d_in[24 + i];
  float* out = (float*)d_out;

  // workspace carve-up
  float* ws   = (float*)d_ws;
  float* Ka   = ws;                                   // H*L
  float* Kb   = Ka   + (size_t)HCH * LSQ;             // H*L
  float* dtar = Kb   + (size_t)HCH * LSQ;             // H*N x4
  float* dtai = dtar + (size_t)HCH * NST;
  float* cbr  = dtai + (size_t)HCH * NST;
  float* cbi  = cbr  + (size_t)HCH * NST;
  float* dvec = cbi  + (size_t)HCH * NST;             // B*C
  float* bufC = dvec + (size_t)BSZ * CCH;             // B*C*L (y0, later gated)
  float* bufA = bufC + (size_t)BSZ * CCH * LSQ;       // B*H*L
  float* bufB = bufA + (size_t)BSZ * HCH * LSQ;       // B*H*L

  const dim3 gH(LT32, HCH / 32, BSZ);                 // h-major 2x2 WMMA grids
  const dim3 gB(LT32, BSZ / 32, HCH);                 // S4 Toeplitz 2x2 grids

  // 1) diffusion projection (WMMA)
  diff_proj_wmma<<<dim3(BSZ / 16, CCH / 16), 32, 0, stream>>>(diff_emb, W_dp, b_dp, dvec);
  // 2) y0 = x + d
  {
    const size_t n = (size_t)BSZ * CCH * LSQ;
    add_diff_kernel<<<(unsigned)((n + 255) / 256), 256, 0, stream>>>(x, dvec, bufC);
  }
  // 3) S4D kernels K_a, K_b
  s4_cb_kernel<<<(HCH * NST + 255) / 256, 256, 0, stream>>>(
      s4a[0], s4a[1], s4a[2], s4a[3], s4a[4], dtar, dtai, cbr, cbi);
  s4_gen_kernel<<<(HCH * LSQ + 255) / 256, 256, 0, stream>>>(dtar, dtai, cbr, cbi, Ka);
  s4_cb_kernel<<<(HCH * NST + 255) / 256, 256, 0, stream>>>(
      s4b[0], s4b[1], s4b[2], s4b[3], s4b[4], dtar, dtai, cbr, cbi);
  s4_gen_kernel<<<(HCH * LSQ + 255) / 256, 256, 0, stream>>>(dtar, dtai, cbr, cbi, Kb);
  // 4) conv k=3 (WMMA 2x2)  bufC -> bufA
  conv3_wmma<<<gH, 32, 0, stream>>>(bufC, W_conv, b_conv, bufA);
  // 5) S4a Toeplitz conv + residual (WMMA 2x2, LDS kernel row)  bufA -> bufB
  s4conv_wmma<<<gB, 32, 0, stream>>>(bufA, Ka, s4a[5], bufB);
  // 6) LayerNorm a (in place)
  layernorm_kernel<<<BSZ * LSQ, 256, 0, stream>>>(bufB, s4a[6], s4a[7]);
  // 7) side + time conditioning (WMMA 2x2, in place)
  cond_add_wmma<<<gH, 32, 0, stream>>>(bufB, cond_obs, cond_mask, time_emb,
                                       W_cond, b_cond, W_time, b_time);
  // 8) S4b Toeplitz conv + residual (WMMA 2x2)  bufB -> bufA
  s4conv_wmma<<<gB, 32, 0, stream>>>(bufB, Kb, s4b[5], bufA);
  // 9) LayerNorm b (in place)
  layernorm_kernel<<<BSZ * LSQ, 256, 0, stream>>>(bufA, s4b[6], s4b[7]);
  // 10) gate  bufA -> bufC
  {
    const size_t n = (size_t)BSZ * CCH * LSQ;
    gate_kernel<<<(unsigned)((n + 255) / 256), 256, 0, stream>>>(bufA, bufC);
  }
  // 11) output projection + residual/skip split (WMMA 2x2)
  out_proj_wmma<<<gH, 32, 0, stream>>>(bufC, W_out, b_out, x, out);
}